// LightGCN_SSL_85598698209628
// MI455X (gfx1250) — compile-verified
//
#include <hip/hip_runtime.h>
#include <hip/hip_bf16.h>

// ---------------------------------------------------------------------------
// LightGCN-SSL propagation for MI455X (gfx1250).
// Memory/atomic-bound COO SpMM. CDNA5 paths used:
//   * global_load_async_to_lds_b128 + s_wait_asynccnt: 4-deep LDS ring of
//     gather stages (ASYNCcnt completes in order -> wait<=3 releases oldest)
//   * nontemporal hints so the three 64MB atomic tables (=192MB = L2 size)
//     stay resident while streaming passes bypass L2
//   * global_prefetch_b8 on the COO index streams
//   * native f32 global atomics (unsafeAtomicAdd -> global_atomic_add_f32)
// WMMA is inapplicable: random COO scatter/gather has no dense tiles and
// only ~5 GFLOP total; the limiter is L2 gather/atomic throughput.
// ---------------------------------------------------------------------------

typedef float v4f __attribute__((ext_vector_type(4)));

namespace lgcn {
constexpr int N_USER = 100000;
constexpr int N_ITEM = 150000;
constexpr int N_NODE = N_USER + N_ITEM;   // 250000
constexpr int D      = 64;
constexpr int LAYERS = 3;
constexpr int B_SZ   = 4096;
constexpr long long BUF = (long long)N_NODE * D;     // 16,000,000 floats (64 MB)
}
using namespace lgcn;

// ---------------------------------------------------------------------------
// init: e_cur = acc = acc1 = acc2 = concat(user_embed, item_embed); e_next = 0
// Embedding sources are read-once -> NT loads. acc is streaming-updated ->
// NT store. e_cur (gather source), e_next/acc1/acc2 (atomic dests) stay RT.
// ---------------------------------------------------------------------------
__global__ __launch_bounds__(256)
void init_buffers(const float* __restrict__ uemb, const float* __restrict__ iemb,
                  float* __restrict__ e_cur, float* __restrict__ e_next,
                  float* __restrict__ acc, float* __restrict__ acc1,
                  float* __restrict__ acc2)
{
    const long long n4 = BUF >> 2;                       // float4 count
    long long i = (long long)blockIdx.x * blockDim.x + threadIdx.x;
    if (i >= n4) return;
    const long long user4 = (long long)N_USER * D / 4;   // 1,600,000
    v4f v = (i < user4) ? __builtin_nontemporal_load((const v4f*)uemb + i)
                        : __builtin_nontemporal_load((const v4f*)iemb + (i - user4));
    ((v4f*)e_cur)[i] = v;
    __builtin_nontemporal_store(v, (v4f*)acc + i);
    ((v4f*)acc1)[i] = v;
    ((v4f*)acc2)[i] = v;
    v4f z = {0.f, 0.f, 0.f, 0.f};
    ((v4f*)e_next)[i] = z;
}

// ---------------------------------------------------------------------------
// COO SpMM with accumulation: y[row,:] += val * x[col,:]
// 16 lanes per nnz, one float4 (b128) per lane. Gathers are staged through a
// 4-slot LDS ring via the gfx1250 async-to-LDS path: 3 gathers in flight per
// wave while the oldest is consumed and scattered with f32 atomics.
// ---------------------------------------------------------------------------
__global__ __launch_bounds__(256)
void spmm_coo(const int*   __restrict__ rows,
              const int*   __restrict__ cols,
              const float* __restrict__ vals,
              long long nnz,
              const float* __restrict__ x,
              float*       __restrict__ y)
{
    constexpr int PIPE = 4;                               // LDS ring depth
    __shared__ v4f stage[PIPE][256];
    const int tid = threadIdx.x;

    const long long WORK   = nnz << 4;                    // 16 work-items / nnz
    const long long stride = (long long)gridDim.x * blockDim.x;  // multiple of 16
    const long long t      = (long long)blockIdx.x * blockDim.x + tid;
    const long long iters  = (WORK + stride - 1) / stride;
    const long long itersPad = (iters + PIPE - 1) & ~(long long)(PIPE - 1);

    const int c4 = (int)(t & 15);                         // invariant per thread

    int   R[PIPE];
    int   C[PIPE];
    float V[PIPE];

    // load (clamped -> always-valid) COO indices for iteration k into a slot
    auto load_idx = [&](long long k, int slot) {
        long long e = (t + k * stride) >> 4;
        if (e >= nnz) e = nnz - 1;
        R[slot] = rows[e]; C[slot] = cols[e]; V[slot] = vals[e];
    };
    // issue async gather of x[C[slot]] chunk c4 into stage[slot][tid]
    auto issue = [&](int slot) {
        const v4f* g = (const v4f*)(x + (long long)C[slot] * D) + c4;
        unsigned lds = (unsigned)(uintptr_t)&stage[slot][tid];
        asm volatile("global_load_async_to_lds_b128 %0, %1, off"
                     :: "v"(lds), "v"(g) : "memory");
    };

    // Prologue: fill PIPE-1 stages (slot = k & 3).
#pragma unroll
    for (int j = 0; j < PIPE - 1; ++j) { load_idx(j, j); issue(j); }

    for (long long kb = 0; kb < itersPad; kb += PIPE) {
        // Stream-prefetch index arrays ~PIPE+2 iterations ahead (1 lane/nnz).
        if (c4 == 0) {
            const long long pf = ((t + kb * stride) >> 4)
                               + (stride >> 4) * (PIPE + 2);
            if (pf < nnz) {
                __builtin_prefetch(rows + pf, 0, 1);
                __builtin_prefetch(cols + pf, 0, 1);
                __builtin_prefetch(vals + pf, 0, 1);
            }
        }
#pragma unroll
        for (int j = 0; j < PIPE; ++j) {                  // k = kb + j, slot = j
            const long long k = kb + j;
            const int slotNext = (j + PIPE - 1) & (PIPE - 1);   // (k+3) & 3
            load_idx(k + PIPE - 1, slotNext);             // slot freed at j-1
            issue(slotNext);
            // In-order ASYNCcnt: <=3 outstanding => stage for k has landed.
            asm volatile("s_wait_asynccnt 0x3" ::: "memory");

            const long long idx = t + k * stride;
            if (idx < WORK) {
                const v4f  xv = stage[j][tid];
                const float v = V[j];
                float* yr = y + (long long)R[j] * D + (long long)c4 * 4;
                unsafeAtomicAdd(yr + 0, v * xv.x);
                unsafeAtomicAdd(yr + 1, v * xv.y);
                unsafeAtomicAdd(yr + 2, v * xv.z);
                unsafeAtomicAdd(yr + 3, v * xv.w);
            }
        }
    }
}

// ---------------------------------------------------------------------------
// acc += e_new (streaming, NT both ways); zero the retiring ping-pong buffer
// (RT store: it is the next layer's atomic destination, keep zeros in L2).
// ---------------------------------------------------------------------------
__global__ __launch_bounds__(256)
void add_and_zero(float* __restrict__ acc, const float* __restrict__ enew,
                  float* __restrict__ eold)
{
    const long long n4 = BUF >> 2;
    long long i = (long long)blockIdx.x * blockDim.x + threadIdx.x;
    if (i >= n4) return;
    v4f a = __builtin_nontemporal_load((const v4f*)acc + i);
    v4f b = ((const v4f*)enew)[i];
    a += b;
    __builtin_nontemporal_store(a, (v4f*)acc + i);
    v4f z = {0.f, 0.f, 0.f, 0.f};
    ((v4f*)eold)[i] = z;
}

// ---------------------------------------------------------------------------
// Final gather + 1/(LAYERS+1) scale into the 9-section output:
//   [ue | pe | ne | ue1 | fi1(pos) | fi1(neg) | ue2 | fi2(pos) | fi2(neg)]
// Output is write-once -> NT stores.
// ---------------------------------------------------------------------------
__global__ __launch_bounds__(256)
void gather_out(const float* __restrict__ acc, const float* __restrict__ acc1,
                const float* __restrict__ acc2,
                const int* __restrict__ user, const int* __restrict__ pos,
                const int* __restrict__ neg, float* __restrict__ out)
{
    const int TOTAL = 9 * B_SZ * 16;                      // 16 lanes per row
    int t = blockIdx.x * blockDim.x + threadIdx.x;
    if (t >= TOTAL) return;
    const int r   = t >> 4;
    const int c4  = t & 15;
    const int sec = r / B_SZ;
    const int j   = r - sec * B_SZ;

    const float* src;
    int idx;
    switch (sec) {
        case 0:  src = acc;  idx = user[j];           break;
        case 1:  src = acc;  idx = N_USER + pos[j];   break;
        case 2:  src = acc;  idx = N_USER + neg[j];   break;
        case 3:  src = acc1; idx = user[j];           break;
        case 4:  src = acc1; idx = N_USER + pos[j];   break;
        case 5:  src = acc1; idx = N_USER + neg[j];   break;
        case 6:  src = acc2; idx = user[j];           break;
        case 7:  src = acc2; idx = N_USER + pos[j];   break;
        default: src = acc2; idx = N_USER + neg[j];   break;
    }
    v4f v = ((const v4f*)(src + (long long)idx * D))[c4];
    const float s = 1.0f / (LAYERS + 1);
    v *= s;
    __builtin_nontemporal_store(v, (v4f*)(out + (long long)r * D) + c4);
}

// ---------------------------------------------------------------------------
// Host side: all launches on `stream`; ws layout = 5 contiguous 64MB tables.
// Every workspace byte used is rewritten each call (deterministic).
// ---------------------------------------------------------------------------
extern "C" void kernel_launch(void* const* d_in, const int* in_sizes, int n_in,
                              void* d_out, int out_size, void* d_ws, size_t ws_size,
                              hipStream_t stream)
{
    const int*   user    = (const int*)  d_in[0];
    const int*   pos     = (const int*)  d_in[1];
    const int*   neg     = (const int*)  d_in[2];
    const int*   A_rows  = (const int*)  d_in[3];
    const int*   A_cols  = (const int*)  d_in[4];
    const float* A_vals  = (const float*)d_in[5];
    const int*   g1_rows = (const int*)  d_in[6];
    const int*   g1_cols = (const int*)  d_in[7];
    const float* g1_vals = (const float*)d_in[8];
    const int*   g2_rows = (const int*)  d_in[9];
    const int*   g2_cols = (const int*)  d_in[10];
    const float* g2_vals = (const float*)d_in[11];
    const float* uemb    = (const float*)d_in[12];
    const float* iemb    = (const float*)d_in[13];

    const long long nnzA = in_sizes[3];
    const long long nnzS = in_sizes[6];

    float* ws    = (float*)d_ws;
    float* e_cur = ws + 0 * BUF;
    float* e_next= ws + 1 * BUF;
    float* acc   = ws + 2 * BUF;
    float* acc1  = ws + 3 * BUF;
    float* acc2  = ws + 4 * BUF;

    const int T = 256;
    const long long n4 = BUF >> 2;
    const int elemGrid = (int)((n4 + T - 1) / T);
    const int spmmGrid = 8192;   // grid-stride; ~40 iters/thread on A

    init_buffers<<<elemGrid, T, 0, stream>>>(uemb, iemb, e_cur, e_next,
                                             acc, acc1, acc2);

    for (int l = 0; l < LAYERS; ++l) {
        // e_next += A * e_cur      (e_next enters zeroed)
        spmm_coo<<<spmmGrid, T, 0, stream>>>(A_rows, A_cols, A_vals, nnzA,
                                             e_cur, e_next);
        // acc += e_next; zero e_cur (becomes next layer's e_next)
        add_and_zero<<<elemGrid, T, 0, stream>>>(acc, e_next, e_cur);
        // acc1 += g1 * e_next ; acc2 += g2 * e_next  (fused accumulation)
        spmm_coo<<<spmmGrid, T, 0, stream>>>(g1_rows, g1_cols, g1_vals, nnzS,
                                             e_next, acc1);
        spmm_coo<<<spmmGrid, T, 0, stream>>>(g2_rows, g2_cols, g2_vals, nnzS,
                                             e_next, acc2);
        float* tmp = e_cur; e_cur = e_next; e_next = tmp;
    }

    const int gatherGrid = (9 * B_SZ * 16 + T - 1) / T;
    gather_out<<<gatherGrid, T, 0, stream>>>(acc, acc1, acc2,
                                             user, pos, neg, (float*)d_out);
}